// GlobalRankPooling_58265526338037
// MI455X (gfx1250) — compile-verified
//
#include <hip/hip_runtime.h>
#include <hip/hip_bf16.h>

typedef __attribute__((ext_vector_type(2))) float v2f;
typedef __attribute__((ext_vector_type(8))) float v8f;

#define B_DIM 32
#define C_DIM 2048
#define S_DIM 1024
#define WAVES_PER_BLK 8
#define XS_STRIDE 36   // 32 floats + 4 pad words -> conflict-free LDS rows

__global__ __launch_bounds__(256) void grp_kernel(const float* __restrict__ x,
                                                  const float* __restrict__ wgt,
                                                  const float* __restrict__ bias,
                                                  float* __restrict__ out) {
    __shared__ float sxs[WAVES_PER_BLK * 32 * XS_STRIDE]; // 36 KB: sorted rows, padded
    __shared__ float swt[WAVES_PER_BLK * S_DIM];          // 32 KB: weight rows

    const int lane = threadIdx.x & 31;
    const int wave = threadIdx.x >> 5;
    const int row  = blockIdx.x * WAVES_PER_BLK + wave;   // row = b*C + c
    const int cidx = row & (C_DIM - 1);

    // ---- kick off async copy of weight row into this wave's LDS slice ----
    const float* wrow = wgt + (size_t)cidx * S_DIM;
    {
        unsigned wslice = (unsigned)(size_t)(swt + wave * S_DIM);
#pragma unroll
        for (int i = 0; i < 8; ++i) {
            unsigned goff  = (unsigned)(lane * 16 + i * 512); // bytes
            unsigned laddr = wslice + goff;
            asm volatile("global_load_async_to_lds_b128 %0, %1, %2 offset:0"
                         :: "v"(laddr), "v"(goff), "s"(wrow) : "memory");
        }
    }

    // ---- load 32 consecutive floats per lane: element e = lane*32 + r ----
    float v[32];
    {
        const float4* xp = (const float4*)(x + (size_t)row * S_DIM + lane * 32);
#pragma unroll
        for (int i = 0; i < 8; ++i) {
            float4 t = xp[i];
            v[4 * i + 0] = t.x; v[4 * i + 1] = t.y;
            v[4 * i + 2] = t.z; v[4 * i + 3] = t.w;
        }
    }

    // ---- bitonic sort, descending, over e = lane*32 + r (1024 elements) ----
    // stage k: block size 2^k; desc iff bit k of e is 0.
    // bits[4:0] of e = r (in-register), bits[9:5] = lane (cross-lane shuffles).
#pragma unroll
    for (int k = 1; k <= 10; ++k) {
#pragma unroll
        for (int j = 512; j >= 1; j >>= 1) {
            if (j > (1 << (k - 1))) continue;
            if (j >= 32) {
                const int m = j >> 5;                       // lane xor mask
                const bool desc   = (((lane >> (k - 5)) & 1) == 0);
                const bool upper  = (lane & m) != 0;
                const bool keepMax = (desc != upper);
#pragma unroll
                for (int r = 0; r < 32; ++r) {
                    float u = __shfl_xor(v[r], m, 32);
                    v[r] = keepMax ? fmaxf(v[r], u) : fminf(v[r], u);
                }
            } else {
#pragma unroll
                for (int r = 0; r < 32; ++r) {
                    if ((r & j) == 0) {
                        const int p = r | j;
                        const bool desc = (k >= 5) ? (((lane >> (k - 5)) & 1) == 0)
                                                   : (((r >> k) & 1) == 0);
                        float a = v[r], b = v[p];
                        bool sw = desc ? (a < b) : (a > b);
                        v[r] = sw ? b : a;
                        v[p] = sw ? a : b;
                    }
                }
            }
        }
    }

    // ---- stash sorted row in padded LDS slice (for WMMA-layout regather) ----
    {
        float4* srow = (float4*)(sxs + wave * 32 * XS_STRIDE + lane * XS_STRIDE);
#pragma unroll
        for (int i = 0; i < 8; ++i)
            srow[i] = make_float4(v[4 * i + 0], v[4 * i + 1], v[4 * i + 2], v[4 * i + 3]);
    }

    // weight row must be resident in LDS before the gather
    asm volatile("s_wait_asynccnt 0x0" ::: "memory");

    // ---- dot(xs_sorted, w) via V_WMMA_F32_16X16X4_F32 ----
    // chunk t (64 elems): A[m,k]=xs[64t+4m+k], B[k,n]=w[64t+4n+k]; diag(C) sums the dot.
    // A frag: VGPR0 = K=(lane>>4)*2, VGPR1 = K+1, M = lane&15 (same formula for B).
    const float* myx = sxs + wave * 32 * XS_STRIDE;
    const float* myw = swt + wave * S_DIM;
    const int midx = lane & 15;
    const int k0   = (lane >> 4) * 2;

    v8f acc = {};
#pragma unroll
    for (int t = 0; t < 16; ++t) {
        const int e = t * 64 + 4 * midx + k0;              // even -> 8B aligned
        v2f a = *(const v2f*)(myx + (e >> 5) * XS_STRIDE + (e & 31));
        v2f b = *(const v2f*)(myw + e);
        acc = __builtin_amdgcn_wmma_f32_16x16x4_f32(
            false, a, false, b, (short)0, acc, false, false);
    }

    // ---- extract diagonal C[m,m] and reduce across the wave ----
    // VGPR d: lanes 0-15 -> (M=d, N=lane); lanes 16-31 -> (M=d+8, N=lane-16)
    float diag = 0.0f;
#pragma unroll
    for (int d = 0; d < 8; ++d)
        diag += ((lane == d) || (lane == 24 + d)) ? acc[d] : 0.0f;
#pragma unroll
    for (int off = 16; off >= 1; off >>= 1)
        diag += __shfl_xor(diag, off, 32);

    if (lane == 0)
        out[row] = diag + bias[cidx];
}

extern "C" void kernel_launch(void* const* d_in, const int* in_sizes, int n_in,
                              void* d_out, int out_size, void* d_ws, size_t ws_size,
                              hipStream_t stream) {
    const float* x    = (const float*)d_in[0];
    const float* wgt  = (const float*)d_in[1];
    const float* bias = (const float*)d_in[2];
    float* out        = (float*)d_out;

    const int rows   = B_DIM * C_DIM;                 // 65536
    const int blocks = rows / WAVES_PER_BLK;          // 8192
    grp_kernel<<<blocks, 256, 0, stream>>>(x, wgt, bias, out);
}